// GNNModel_8615704396307
// MI455X (gfx1250) — compile-verified
//
#include <hip/hip_runtime.h>

#define B_G   64
#define NPG   1024
#define KNN   20
#define HD    16
#define NL    4
#define NTOT  (B_G * NPG)
#define ETOT  (NTOT * KNN)

#define NPB   32              // nodes per block in layer kernel
#define EPB   (NPB * KNN)     // 640 edges per block
#define TPB   128             // 4 waves of 32
#define NWAVE 4
#define TILES (EPB / 16)      // 40 edge tiles
#define TILES_PER_WAVE (TILES / NWAVE)

typedef float v2f __attribute__((ext_vector_type(2)));
typedef float v8f __attribute__((ext_vector_type(8)));

__device__ __forceinline__ float silu_f(float x) { return x / (1.f + __expf(-x)); }

// Same-wave LDS producer->consumer fence (LDS is in-order per wave; wait for stores).
__device__ __forceinline__ void wave_lds_fence() {
  asm volatile("s_wait_dscnt 0" ::: "memory");
}

// C/D tile init: C[m][n] = bias[n0 + n]; in C layout every VGPR holds col n = lane&15.
__device__ __forceinline__ v8f bias_tile(const float* b, int n0) {
  float bv = b[n0 + (threadIdx.x & 15)];
  v8f c;
#pragma unroll
  for (int r = 0; r < 8; ++r) c[r] = bv;
  return c;
}

// D += ACT(16xKD, row-major ldA) * W(KDxN, row-major ldW, col offset n0) via f32 WMMA.
// A layout: A[m][k]: vgpr=k&1, lane=(k>>1)*16+m.  B assumed mirrored: B[k][n]: vgpr=k&1, lane=(k>>1)*16+n.
template <int KD>
__device__ __forceinline__ v8f mm_tile(const float* act, int ldA,
                                       const float* w, int ldW, int n0, v8f c) {
  const int lane = threadIdx.x & 31;
  const int half = lane >> 4;
  const int lpos = lane & 15;
#pragma unroll
  for (int k0 = 0; k0 < KD; k0 += 4) {
    const int ka = k0 + (half << 1);
    v2f a, b;
    a.x = act[lpos * ldA + ka];
    a.y = act[lpos * ldA + ka + 1];
    b.x = w[ka * ldW + n0 + lpos];
    b.y = w[(ka + 1) * ldW + n0 + lpos];
    c = __builtin_amdgcn_wmma_f32_16x16x4_f32(false, a, false, b, (short)0, c, false, false);
  }
  return c;
}

// Scatter a 16x16 C tile to LDS (optionally through SiLU). C[m][n]: vgpr=m&7, lane=(m>=8)*16+n.
__device__ __forceinline__ void store_tile(float* out, int ldO, v8f c, bool act_silu) {
  const int lane = threadIdx.x & 31;
  const int half = lane >> 4;
  const int n = lane & 15;
#pragma unroll
  for (int r = 0; r < 8; ++r) {
    const int m = half * 8 + r;
    float v = c[r];
    if (act_silu) v = silu_f(v);
    out[m * ldO + n] = v;
  }
}

// ---------------- kNN + geometric edge features ----------------
__global__ __launch_bounds__(256) void knn_geo_kernel(const float* __restrict__ pos,
                                                      int* __restrict__ nbr,
                                                      float* __restrict__ geo) {
  __shared__ float px[NPG], py[NPG], pz[NPG];
  const int g = blockIdx.x;
  const float* gp = pos + (size_t)g * NPG * 3;
  for (int i = threadIdx.x; i < NPG; i += blockDim.x) {
    px[i] = gp[3 * i]; py[i] = gp[3 * i + 1]; pz[i] = gp[3 * i + 2];
  }
  __syncthreads();
  const float INF = __builtin_inff();
  for (int i = threadIdx.x; i < NPG; i += blockDim.x) {
    const float xi = px[i], yi = py[i], zi = pz[i];
    float bd[KNN]; int bi[KNN];
#pragma unroll
    for (int k = 0; k < KNN; ++k) { bd[k] = INF; bi[k] = 0; }
    for (int j = 0; j < NPG; ++j) {
      float dx = px[j] - xi, dy = py[j] - yi, dz = pz[j] - zi;
      float d2 = dx * dx + dy * dy + dz * dz;
      if (j == i) d2 = INF;
      if (d2 < bd[KNN - 1]) {
        float cd = d2; int ci = j;
#pragma unroll
        for (int k = 0; k < KNN; ++k) {
          if (cd < bd[k]) { float td = bd[k]; int ti = bi[k]; bd[k] = cd; bi[k] = ci; cd = td; ci = ti; }
        }
      }
    }
    const float rni = 1.f / (sqrtf(xi * xi + yi * yi + zi * zi) + 1e-8f);
    const size_t eb = ((size_t)g * NPG + i) * KNN;
#pragma unroll
    for (int k = 0; k < KNN; ++k) {
      const int j = bi[k];
      nbr[eb + k] = g * NPG + j;
      const float xj = px[j], yj = py[j], zj = pz[j];
      const float rx = xj - xi, ry = yj - yi, rz = zj - zi;
      const float dist = sqrtf(rx * rx + ry * ry + rz * rz);
      const float rd = 1.f / (dist + 1e-8f);
      const float rnj = 1.f / (sqrtf(xj * xj + yj * yj + zj * zj) + 1e-8f);
      const float dot = (xi * xj + yi * yj + zi * zj) * rni * rnj;
      float* go = geo + (eb + k) * 5;
      go[0] = dist; go[1] = rx * rd; go[2] = ry * rd; go[3] = rz * rd; go[4] = dot;
    }
  }
}

// ---------------- node encoder: h0 = silu(pos@W1+b1)@W2+b2 ----------------
__global__ __launch_bounds__(128) void encode_kernel(const float* __restrict__ pos,
    const float* __restrict__ w1, const float* __restrict__ b1,
    const float* __restrict__ w2, const float* __restrict__ b2,
    float* __restrict__ h) {
  const int i = blockIdx.x * blockDim.x + threadIdx.x;
  if (i >= NTOT) return;
  const float x = pos[3 * i], y = pos[3 * i + 1], z = pos[3 * i + 2];
  float t[HD];
#pragma unroll
  for (int n = 0; n < HD; ++n)
    t[n] = silu_f(x * w1[n] + y * w1[HD + n] + z * w1[2 * HD + n] + b1[n]);
#pragma unroll
  for (int n = 0; n < HD; ++n) {
    float s = b2[n];
#pragma unroll
    for (int k = 0; k < HD; ++k) s += t[k] * w2[k * HD + n];
    h[(size_t)i * HD + n] = s;
  }
}

// ---------------- one message-passing layer (fused edge MLP + agg + node MLP) ----------------
__global__ __launch_bounds__(TPB) void layer_kernel(
    const float* __restrict__ hin, float* __restrict__ hout,
    const int* __restrict__ nbr, const float* __restrict__ geo,
    const float* __restrict__ ew1, const float* __restrict__ eb1,
    const float* __restrict__ ew2, const float* __restrict__ eb2,
    const float* __restrict__ ew3, const float* __restrict__ eb3,
    const float* __restrict__ nw1, const float* __restrict__ nb1,
    const float* __restrict__ nw2, const float* __restrict__ nb2,
    int l) {
  __shared__ float sm[EPB * HD];          // per-edge messages m[640][16]
  __shared__ float sef[NWAVE * 16 * 40];  // per-wave activation staging [16][40]
  __shared__ float scf[NPB * 48];         // [h, mean, max] per node
  __shared__ float sw1[40 * 32];          // ew1 padded 37->40 rows
  __shared__ float sw2[32 * 16];
  __shared__ float sw3[16 * 16];
  __shared__ float snw1[48 * 32];
  __shared__ float snw2[32 * 16];
  __shared__ float sb1[32], sb2[16], sb3[16], snb1[32], snb2[16];

  const int tid  = threadIdx.x;
  const int wave = tid >> 5;
  const int lane = tid & 31;
  const int half = lane >> 4;
  const int lpos = lane & 15;
  const int blk_n0 = blockIdx.x * NPB;
  const size_t blk_e0 = (size_t)blockIdx.x * EPB;

  // stage this layer's weights/biases into LDS
  {
    const float* w = ew1 + (size_t)l * 37 * 32;
    for (int idx = tid; idx < 40 * 32; idx += TPB) {
      const int k = idx >> 5;
      sw1[idx] = (k < 37) ? w[idx] : 0.f;
    }
    const float* p2 = ew2 + (size_t)l * 32 * 16;
    for (int idx = tid; idx < 32 * 16; idx += TPB) sw2[idx] = p2[idx];
    const float* p3 = ew3 + (size_t)l * 16 * 16;
    for (int idx = tid; idx < 16 * 16; idx += TPB) sw3[idx] = p3[idx];
    const float* q1 = nw1 + (size_t)l * 48 * 32;
    for (int idx = tid; idx < 48 * 32; idx += TPB) snw1[idx] = q1[idx];
    const float* q2 = nw2 + (size_t)l * 32 * 16;
    for (int idx = tid; idx < 32 * 16; idx += TPB) snw2[idx] = q2[idx];
    if (tid < 32) { sb1[tid] = eb1[l * 32 + tid]; snb1[tid] = nb1[l * 32 + tid]; }
    if (tid < 16) { sb2[tid] = eb2[l * 16 + tid]; sb3[tid] = eb3[l * 16 + tid]; snb2[tid] = nb2[l * 16 + tid]; }
  }
  __syncthreads();

  // ---- edge phase: each wave runs 10 tiles of 16 edges through the 3-layer edge MLP ----
  float* ef = sef + wave * (16 * 40);
  for (int it = 0; it < TILES_PER_WAVE; ++it) {
    const int t = wave * TILES_PER_WAVE + it;
    const int le = t * 16 + lpos;          // local edge index
    const size_t ge = blk_e0 + le;         // global edge index
    const int i_node = blk_n0 + le / KNN;  // dst node (edges contiguous per node)
    float* row = ef + lpos * 40;
    if (half == 0) {  // h_i -> cols 0..15, geo -> 32..36, pad 37..39
      const float4* hp = (const float4*)(hin + (size_t)i_node * HD);
#pragma unroll
      for (int q = 0; q < 4; ++q) ((float4*)row)[q] = hp[q];
      const float* gp = geo + ge * 5;
      row[32] = gp[0]; row[33] = gp[1]; row[34] = gp[2]; row[35] = gp[3]; row[36] = gp[4];
      row[37] = 0.f; row[38] = 0.f; row[39] = 0.f;
    } else {          // h_j -> cols 16..31
      const int src = nbr[ge];
      const float4* hp = (const float4*)(hin + (size_t)src * HD);
#pragma unroll
      for (int q = 0; q < 4; ++q) ((float4*)(row + 16))[q] = hp[q];
    }
    wave_lds_fence();
    // L1: [16x40] @ [40x32] + b, SiLU
    v8f cA = mm_tile<40>(ef, 40, sw1, 32, 0,  bias_tile(sb1, 0));
    v8f cB = mm_tile<40>(ef, 40, sw1, 32, 16, bias_tile(sb1, 16));
    store_tile(ef,      40, cA, true);
    store_tile(ef + 16, 40, cB, true);
    wave_lds_fence();
    // L2: [16x32] @ [32x16] + b, SiLU
    v8f c2 = mm_tile<32>(ef, 40, sw2, 16, 0, bias_tile(sb2, 0));
    store_tile(ef, 40, c2, true);
    wave_lds_fence();
    // L3: [16x16] @ [16x16] + b  -> messages
    v8f c3 = mm_tile<16>(ef, 40, sw3, 16, 0, bias_tile(sb3, 0));
    store_tile(sm + t * 16 * HD, HD, c3, false);
  }
  __syncthreads();

  // ---- aggregation: contiguous 20-edge segments -> mean & max, plus copy h ----
  for (int p = tid; p < NPB * HD; p += TPB) {
    const int ln = p >> 4;
    const int n  = p & 15;
    float s = 0.f, mx = -__builtin_inff();
#pragma unroll
    for (int e = 0; e < KNN; ++e) {
      const float v = sm[(ln * KNN + e) * HD + n];
      s += v;
      mx = fmaxf(mx, v);
    }
    scf[ln * 48 + 16 + n] = s * (1.f / (float)KNN);
    scf[ln * 48 + 32 + n] = mx;
    scf[ln * 48 + n] = hin[(size_t)(blk_n0 + ln) * HD + n];
  }
  __syncthreads();

  // ---- node phase: upd = silu(cf@nw1+nb1)@nw2+nb2 ; hout = h + upd ----
  if (wave < 2) {
    const float* act = scf + wave * 16 * 48;
    v8f cA = mm_tile<48>(act, 48, snw1, 32, 0,  bias_tile(snb1, 0));
    v8f cB = mm_tile<48>(act, 48, snw1, 32, 16, bias_tile(snb1, 16));
    float* st = sef + wave * (16 * 40);
    store_tile(st,      40, cA, true);
    store_tile(st + 16, 40, cB, true);
    wave_lds_fence();
    v8f c2 = mm_tile<32>(st, 40, snw2, 16, 0, bias_tile(snb2, 0));
#pragma unroll
    for (int r = 0; r < 8; ++r) {
      const int m = half * 8 + r;
      const float v = c2[r] + scf[(wave * 16 + m) * 48 + lpos];
      hout[(size_t)(blk_n0 + wave * 16 + m) * HD + lpos] = v;
    }
  }
}

// ---------------- output head ----------------
__global__ __launch_bounds__(128) void head_kernel(const float* __restrict__ h,
    const float* __restrict__ ow1, const float* __restrict__ ob1,
    const float* __restrict__ ow2, const float* __restrict__ ob2,
    const float* __restrict__ ow3, const float* __restrict__ ob3,
    float* __restrict__ out) {
  const int i = blockIdx.x * blockDim.x + threadIdx.x;
  if (i >= NTOT) return;
  float hv[16];
#pragma unroll
  for (int k = 0; k < 16; ++k) hv[k] = h[(size_t)i * 16 + k];
  float a[16];
#pragma unroll
  for (int n = 0; n < 16; ++n) {
    float s = ob1[n];
#pragma unroll
    for (int k = 0; k < 16; ++k) s += hv[k] * ow1[k * 16 + n];
    a[n] = silu_f(s);
  }
  float b[8];
#pragma unroll
  for (int n = 0; n < 8; ++n) {
    float s = ob2[n];
#pragma unroll
    for (int k = 0; k < 16; ++k) s += a[k] * ow2[k * 8 + n];
    b[n] = silu_f(s);
  }
#pragma unroll
  for (int c = 0; c < 3; ++c) {
    float s = ob3[c];
#pragma unroll
    for (int k = 0; k < 8; ++k) s += b[k] * ow3[k * 3 + c];
    out[(size_t)i * 3 + c] = s;
  }
}

extern "C" void kernel_launch(void* const* d_in, const int* in_sizes, int n_in,
                              void* d_out, int out_size, void* d_ws, size_t ws_size,
                              hipStream_t stream) {
  (void)in_sizes; (void)n_in; (void)out_size; (void)ws_size;
  const float* pos   = (const float*)d_in[0];
  const float* pe_w1 = (const float*)d_in[1];
  const float* pe_b1 = (const float*)d_in[2];
  const float* pe_w2 = (const float*)d_in[3];
  const float* pe_b2 = (const float*)d_in[4];
  const float* ew1   = (const float*)d_in[5];
  const float* eb1   = (const float*)d_in[6];
  const float* ew2   = (const float*)d_in[7];
  const float* eb2   = (const float*)d_in[8];
  const float* ew3   = (const float*)d_in[9];
  const float* eb3   = (const float*)d_in[10];
  const float* nw1   = (const float*)d_in[11];
  const float* nb1   = (const float*)d_in[12];
  const float* nw2   = (const float*)d_in[13];
  const float* nb2   = (const float*)d_in[14];
  const float* ow1   = (const float*)d_in[15];
  const float* ob1   = (const float*)d_in[16];
  const float* ow2   = (const float*)d_in[17];
  const float* ob2   = (const float*)d_in[18];
  const float* ow3   = (const float*)d_in[19];
  const float* ob3   = (const float*)d_in[20];

  char* ws = (char*)d_ws;
  size_t off = 0;
  auto take = [&](size_t bytes) -> char* {
    char* p = ws + off;
    off += (bytes + 255) & ~(size_t)255;
    return p;
  };
  int*   nbr = (int*)  take((size_t)ETOT * sizeof(int));
  float* geo = (float*)take((size_t)ETOT * 5 * sizeof(float));
  float* hA  = (float*)take((size_t)NTOT * HD * sizeof(float));
  float* hB  = (float*)take((size_t)NTOT * HD * sizeof(float));

  knn_geo_kernel<<<B_G, 256, 0, stream>>>(pos, nbr, geo);
  encode_kernel<<<(NTOT + 127) / 128, 128, 0, stream>>>(pos, pe_w1, pe_b1, pe_w2, pe_b2, hA);
  for (int l = 0; l < NL; ++l) {
    const float* hi = (l & 1) ? hB : hA;   // double-buffer: other blocks gather neighbor h
    float*       ho = (l & 1) ? hA : hB;
    layer_kernel<<<NTOT / NPB, TPB, 0, stream>>>(hi, ho, nbr, geo,
        ew1, eb1, ew2, eb2, ew3, eb3, nw1, nb1, nw2, nb2, l);
  }
  head_kernel<<<(NTOT + 127) / 128, 128, 0, stream>>>(hA, ow1, ob1, ow2, ob2, ow3, ob3,
                                                      (float*)d_out);
}